// gLASE_5317169512923
// MI455X (gfx1250) — compile-verified
//
#include <hip/hip_runtime.h>

// gLASE on MI455X (gfx1250): sparse edge-set formulation + fp32 WMMA for all
// D=16 dense algebra. N=4096, D=16, STEPS=3.
//
// Per step s:
//   xq = x @ Q            x2 = x @ W2[s]^T                (WMMA f32 16x16x4)
//   y1[j] += x[i]          over deduped (edge_index ∩ mask) edges (atomics)
//   y2[j] += <xq[i],x[j]> * x2[i]  over (edge_index_2 ∩ mask)     (atomics)
//   x'   = x + (y1 @ (W1[s]^T @ Q)) / (n*p1) - (y2 @ Q) * n/cnt2  (WMMA)

#define NN 4096
#define DD 16

typedef float v2f __attribute__((ext_vector_type(2)));
typedef float v8f __attribute__((ext_vector_type(8)));

// ---------------- WMMA fragment helpers (V_WMMA_F32_16X16X4_F32) -----------
// A 16x4 f32: lanes 0-15 hold M=lane, K={0,1}; lanes 16-31 hold M=lane-16,
// K={2,3}. We chain 4 chunks to get K=16.
__device__ __forceinline__ void load_a_frag(const float* __restrict__ X,
                                            int row0, float scale, v2f a[4]) {
  const int lane = threadIdx.x & 31;
  const int m = lane & 15, h = lane >> 4;
  const float* base = X + (size_t)(row0 + m) * DD + 2 * h;
#pragma unroll
  for (int c = 0; c < 4; ++c) {
    a[c].x = base[4 * c] * scale;
    a[c].y = base[4 * c + 1] * scale;
  }
}

// B 4x16 f32 (K x N), from row-major W[k][n] (B = W)
__device__ __forceinline__ void load_b_frag(const float* __restrict__ W, v2f b[4]) {
  const int lane = threadIdx.x & 31;
  const int n = lane & 15, h = lane >> 4;
#pragma unroll
  for (int c = 0; c < 4; ++c) {
    b[c].x = W[(4 * c + 2 * h) * DD + n];
    b[c].y = W[(4 * c + 2 * h + 1) * DD + n];
  }
}

// B 4x16 f32 from transposed weight: B[k][n] = W[n][k]
__device__ __forceinline__ void load_bT_frag(const float* __restrict__ W, v2f b[4]) {
  const int lane = threadIdx.x & 31;
  const int n = lane & 15, h = lane >> 4;
  const float* row = W + n * DD;
#pragma unroll
  for (int c = 0; c < 4; ++c) {
    b[c].x = row[4 * c + 2 * h];
    b[c].y = row[4 * c + 2 * h + 1];
  }
}

// C/D 16x16 f32: VGPR r holds C[r + 8*(lane>>4)][lane&15]
__device__ __forceinline__ v8f load_c_tile(const float* __restrict__ X, int row0) {
  const int lane = threadIdx.x & 31;
  const int n = lane & 15, h = lane >> 4;
  v8f c;
#pragma unroll
  for (int r = 0; r < 8; ++r) c[r] = X[(size_t)(row0 + r + 8 * h) * DD + n];
  return c;
}

__device__ __forceinline__ void store_d_tile(float* __restrict__ O, int row0, v8f d) {
  const int lane = threadIdx.x & 31;
  const int n = lane & 15, h = lane >> 4;
#pragma unroll
  for (int r = 0; r < 8; ++r) O[(size_t)(row0 + r + 8 * h) * DD + n] = d[r];
}

__device__ __forceinline__ v8f wmma16(const v2f a[4], const v2f b[4], v8f c) {
#pragma unroll
  for (int k = 0; k < 4; ++k)
    c = __builtin_amdgcn_wmma_f32_16x16x4_f32(false, a[k], false, b[k],
                                              (short)0, c, false, false);
  return c;
}

// ---------------- Sparse set construction ----------------------------------
__global__ void k_build_mask(const int* __restrict__ mrow, const int* __restrict__ mcol,
                             int E, unsigned* __restrict__ Md) {
  int e = blockIdx.x * blockDim.x + threadIdx.x;
  if (e >= E) return;
  unsigned bit = (unsigned)mrow[e] * (unsigned)NN + (unsigned)mcol[e];
  atomicOr(&Md[bit >> 5], 1u << (bit & 31));
}

// intersect edge list with mask bitmap, dedup via per-adjacency bitmap,
// compact surviving (i,j) pairs into list (i<<12 | j), count into *cnt.
__global__ void k_build_adj(const int* __restrict__ erow, const int* __restrict__ ecol,
                            int E, const unsigned* __restrict__ Md,
                            unsigned* __restrict__ Ab, unsigned* __restrict__ list,
                            unsigned* __restrict__ cnt) {
  int e = blockIdx.x * blockDim.x + threadIdx.x;
  if (e >= E) return;
  unsigned i = (unsigned)erow[e], j = (unsigned)ecol[e];
  unsigned bit = i * (unsigned)NN + j;
  unsigned w = bit >> 5, b = 1u << (bit & 31);
  if (!(Md[w] & b)) return;                 // not in mask
  unsigned old = atomicOr(&Ab[w], b);
  if (old & b) return;                      // duplicate edge
  unsigned idx = atomicAdd(cnt, 1u);
  list[idx] = (i << 12) | j;
}

// G1[s] = W1[s]^T @ Q   (fuses the (y1 @ W1^T) @ Q chain into one GEMM)
__global__ void k_g1(const float* __restrict__ W1, const float* __restrict__ Q,
                     float* __restrict__ G1) {
  int s = blockIdx.x;
  int k = threadIdx.x >> 4, n = threadIdx.x & 15;
  float acc = 0.f;
#pragma unroll
  for (int t = 0; t < DD; ++t)
    acc = fmaf(W1[s * DD * DD + t * DD + k], Q[t * DD + n], acc);
  G1[s * DD * DD + k * DD + n] = acc;
}

// ---------------- Per-step kernels ------------------------------------------
// xq = x @ Q ; x2 = x @ W2s^T   (one 16-row tile per wave)
__global__ void k_pre(const float* __restrict__ x, const float* __restrict__ Qm,
                      const float* __restrict__ W2s, float* __restrict__ xq,
                      float* __restrict__ x2) {
  int wave = threadIdx.x >> 5;
  int row0 = (blockIdx.x * (blockDim.x >> 5) + wave) * 16;
  v2f a[4], b[4];
  load_a_frag(x, row0, 1.0f, a);
  load_b_frag(Qm, b);
  v8f d = {};
  d = wmma16(a, b, d);
  store_d_tile(xq, row0, d);
  load_bT_frag(W2s, b);
  v8f d2 = {};
  d2 = wmma16(a, b, d2);
  store_d_tile(x2, row0, d2);
}

// y1 = A1^T @ x : for each A1 edge (i,j): y1[j] += x[i]
__global__ void k_scatter1(const unsigned* __restrict__ list,
                           const unsigned* __restrict__ cnt,
                           const float* __restrict__ x, float* __restrict__ y1) {
  unsigned e = blockIdx.x * blockDim.x + threadIdx.x;
  if (e >= *cnt) return;
  unsigned p = list[e];
  const float* xs = x + (size_t)(p >> 12) * DD;
  float* yd = y1 + (size_t)(p & 4095u) * DD;
#pragma unroll
  for (int c = 0; c < DD; ++c) atomicAdd(&yd[c], xs[c]);
}

// y2 = attn^T @ x2 : for each A2 edge (i,j):
//   a = <xq[i], x[j]>;  y2[j] += a * x2[i]
__global__ void k_scatter2(const unsigned* __restrict__ list,
                           const unsigned* __restrict__ cnt,
                           const float* __restrict__ xq, const float* __restrict__ x,
                           const float* __restrict__ x2, float* __restrict__ y2) {
  unsigned e = blockIdx.x * blockDim.x + threadIdx.x;
  if (e >= *cnt) return;
  unsigned p = list[e];
  unsigned i = p >> 12, j = p & 4095u;
  const float* qi = xq + (size_t)i * DD;
  const float* xj = x + (size_t)j * DD;
  float a = 0.f;
#pragma unroll
  for (int c = 0; c < DD; ++c) a = fmaf(qi[c], xj[c], a);
  const float* v = x2 + (size_t)i * DD;
  float* yd = y2 + (size_t)j * DD;
#pragma unroll
  for (int c = 0; c < DD; ++c) atomicAdd(&yd[c], a * v[c]);
}

// x' = x + (y1 @ G1s) * s1 - (y2 @ Q) * (n/cnt2)
// scales folded into A-fragments; both products accumulate into C = x tile.
__global__ void k_combine(const float* __restrict__ xcur, const float* __restrict__ y1,
                          const float* __restrict__ y2, const float* __restrict__ G1s,
                          const float* __restrict__ Qm, const unsigned* __restrict__ cnt2,
                          float s1, float* __restrict__ xnext) {
  int wave = threadIdx.x >> 5;
  int row0 = (blockIdx.x * (blockDim.x >> 5) + wave) * 16;
  float s2 = -(float)NN / (float)(*cnt2);   // negated: subtract via accumulate
  v2f a[4], b[4];
  v8f c = load_c_tile(xcur, row0);
  load_a_frag(y1, row0, s1, a);
  load_b_frag(G1s, b);
  c = wmma16(a, b, c);
  load_a_frag(y2, row0, s2, a);
  load_b_frag(Qm, b);
  c = wmma16(a, b, c);
  store_d_tile(xnext, row0, c);
}

// ---------------- Host driver ------------------------------------------------
extern "C" void kernel_launch(void* const* d_in, const int* in_sizes, int n_in,
                              void* d_out, int out_size, void* d_ws, size_t ws_size,
                              hipStream_t stream) {
  const float* x_in = (const float*)d_in[0];
  const int* e1 = (const int*)d_in[1];
  const int* e2 = (const int*)d_in[2];
  const float* Qm = (const float*)d_in[3];
  const int* mk = (const int*)d_in[4];
  const float* W1 = (const float*)d_in[5];
  const float* W2 = (const float*)d_in[6];
  float* out = (float*)d_out;

  const int E1 = in_sizes[1] / 2;
  const int E2 = in_sizes[2] / 2;
  const int EM = in_sizes[4] / 2;
  const int STEPS = in_sizes[5] / (DD * DD);

  // workspace carve-out (256B aligned regions)
  char* ws = (char*)d_ws;
  size_t off = 0;
  auto carve = [&](size_t bytes) {
    char* p = ws + off;
    off += (bytes + 255) & ~(size_t)255;
    return p;
  };
  const size_t BM = (size_t)NN * NN / 8;          // 2 MB bitmap
  unsigned* Md    = (unsigned*)carve(BM);
  unsigned* A1b   = (unsigned*)carve(BM);
  unsigned* A2b   = (unsigned*)carve(BM);
  unsigned* list1 = (unsigned*)carve((size_t)E1 * 4);
  unsigned* list2 = (unsigned*)carve((size_t)E2 * 4);
  unsigned* cnts  = (unsigned*)carve(8);          // [0]=cnt1 [1]=cnt2
  float* G1  = (float*)carve((size_t)STEPS * DD * DD * 4);
  float* xq  = (float*)carve((size_t)NN * DD * 4);
  float* x2  = (float*)carve((size_t)NN * DD * 4);
  float* y1  = (float*)carve((size_t)NN * DD * 4);
  float* y2  = (float*)carve((size_t)NN * DD * 4);
  float* xb0 = (float*)carve((size_t)NN * DD * 4);
  float* xb1 = (float*)carve((size_t)NN * DD * 4);

  // Md/A1b/A2b are contiguous (BM is 256B-multiple) -> one memset
  hipMemsetAsync(Md, 0, 3 * BM, stream);
  hipMemsetAsync(cnts, 0, 8, stream);

  k_build_mask<<<(EM + 255) / 256, 256, 0, stream>>>(mk, mk + EM, EM, Md);
  k_build_adj<<<(E1 + 255) / 256, 256, 0, stream>>>(e1, e1 + E1, E1, Md, A1b, list1, cnts + 0);
  k_build_adj<<<(E2 + 255) / 256, 256, 0, stream>>>(e2, e2 + E2, E2, Md, A2b, list2, cnts + 1);
  k_g1<<<STEPS, 256, 0, stream>>>(W1, Qm, G1);

  const float s1 = (float)NN / (float)EM;   // 1/(n*p1) = n/|mask|

  const int TILE_BLOCKS = NN / 16 / 8;      // 8 waves (tiles) per 256-thread block
  const float* xc = x_in;
  for (int s = 0; s < STEPS; ++s) {
    float* xn = (s == STEPS - 1) ? out : ((s & 1) ? xb1 : xb0);
    hipMemsetAsync(y1, 0, (size_t)NN * DD * 4, stream);
    hipMemsetAsync(y2, 0, (size_t)NN * DD * 4, stream);
    k_pre<<<TILE_BLOCKS, 256, 0, stream>>>(xc, Qm, W2 + (size_t)s * DD * DD, xq, x2);
    k_scatter1<<<(E1 + 255) / 256, 256, 0, stream>>>(list1, cnts + 0, xc, y1);
    k_scatter2<<<(E2 + 255) / 256, 256, 0, stream>>>(list2, cnts + 1, xq, xc, x2, y2);
    k_combine<<<TILE_BLOCKS, 256, 0, stream>>>(xc, y1, y2, G1 + (size_t)s * DD * DD,
                                               Qm, cnts + 1, s1, xn);
    xc = xn;
  }
}